// UncoupledLSTM_18854906429607
// MI455X (gfx1250) — compile-verified
//
#include <hip/hip_runtime.h>

// ---------------------------------------------------------------------------
// UncoupledLSTM pipeline, dead-code-eliminated.
//
// The reference returns ONLY the DLT triangulation of frame 0 vs frame T/2;
// the LSTM / MLP / exponential-filter results are never used by the output.
// So the optimal MI455X kernel is a batched 4x4 DLT solve over 544 points:
//   A(4x4) per point  ->  G = A^T A  (via V_WMMA_F32_16X16X4_F32, 4 pts/wave)
//   -> smallest-eigenvalue eigenvector of G via cyclic Jacobi (per lane)
//   -> max-abs normalize, w-guard, xyz = hom[:3]/w  (NaN if |w| < 0.005)
//
// WMMA packing trick: with A-operand rows = blockdiag(A_q^T) and B-operand
// cols = blockdiag(A_q), the per-lane values of the two operands coincide
// (A-op[M=4q+r,k] = A_q[k,r], B-op[k,N=4q+c] = A_q[k,c]), so a single
// 2-VGPR register pair feeds both SRC0 and SRC1 of the WMMA.
// ---------------------------------------------------------------------------

typedef __attribute__((ext_vector_type(2))) float v2f;
typedef __attribute__((ext_vector_type(8))) float v8f;

#define T_SEQ 256
#define J_PTS 17

__global__ __launch_bounds__(256) void tri_dlt_kernel(
    const float* __restrict__ in2d,    // (32, 256, 17, 2)
    const float* __restrict__ incam,   // (32, 256, 3, 4)
    float* __restrict__ out)           // (32, 1, 17, 3)
{
    __shared__ float lds[8 * 256];     // one 16x16 f32 WMMA tile per wave

    const int tid   = threadIdx.x;
    const int lane  = tid & 31;
    const int wave  = tid >> 5;                 // 0..7
    const int waveG = blockIdx.x * 8 + wave;    // 0..135
    const int l15   = lane & 15;
    const int q     = l15 >> 2;                 // local point 0..3
    const int r     = l15 & 3;                  // DLT column 0..3
    const int p     = waveG * 4 + q;            // global point 0..543
    const int b     = p / J_PTS;
    const int j     = p - b * J_PTS;

    // frame 0 and frame T/2 slices
    const int s0 = ((b * T_SEQ + 0)       * J_PTS + j) * 2;
    const int s1 = ((b * T_SEQ + T_SEQ/2) * J_PTS + j) * 2;
    const int c0 = (b * T_SEQ + 0)       * 12;
    const int c1 = (b * T_SEQ + T_SEQ/2) * 12;

    // lanes 0-15 supply DLT rows {0,1} (camera A), lanes 16-31 rows {2,3} (camera B)
    float a0, a1;
    if (lane < 16) {
        const float x  = in2d[s0 + 0];
        const float y  = in2d[s0 + 1];
        const float p2 = incam[c0 + 8 + r];
        a0 = x * p2 - incam[c0 + 0 + r];        // A_q[0, r]
        a1 = y * p2 - incam[c0 + 4 + r];        // A_q[1, r]
    } else {
        const float x  = in2d[s1 + 0];
        const float y  = in2d[s1 + 1];
        const float p2 = incam[c1 + 8 + r];
        a0 = x * p2 - incam[c1 + 0 + r];        // A_q[2, r]
        a1 = y * p2 - incam[c1 + 4 + r];        // A_q[3, r]
    }

    // EXEC is all-ones here (branch reconverged). One WMMA = 4 Gram matrices.
    v2f ab;
    ab.x = a0;
    ab.y = a1;
    v8f acc = {0.f, 0.f, 0.f, 0.f, 0.f, 0.f, 0.f, 0.f};
    acc = __builtin_amdgcn_wmma_f32_16x16x4_f32(
        false, ab, false, ab, (short)0, acc, false, false);

    // Spill C tile: VGPR i holds (M = i or i+8, N = lane or lane-16)
    const int base = wave * 256;
    const int mofs = (lane < 16) ? 0 : 8;
#pragma unroll
    for (int i = 0; i < 8; ++i)
        lds[base + (mofs + i) * 16 + l15] = acc[i];

    __syncthreads();

    // 4 lanes per wave: Jacobi eigensolve of the 4x4 diagonal block
    if (lane < 4) {
        const int qq = lane;
        const int pp = waveG * 4 + qq;

        float G[4][4], V[4][4];
#pragma unroll
        for (int i = 0; i < 4; ++i)
#pragma unroll
            for (int k = 0; k < 4; ++k) {
                G[i][k] = lds[base + (4 * qq + i) * 16 + (4 * qq + k)];
                V[i][k] = (i == k) ? 1.f : 0.f;
            }

        for (int sweep = 0; sweep < 8; ++sweep) {
#pragma unroll
            for (int pi = 0; pi < 3; ++pi) {
#pragma unroll
                for (int qi = pi + 1; qi < 4; ++qi) {
                    const float apq = G[pi][qi];
                    if (fabsf(apq) < 1e-20f) continue;
                    const float tau = (G[qi][qi] - G[pi][pi]) / (2.f * apq);
                    const float t   = (tau >= 0.f ? 1.f : -1.f) /
                                      (fabsf(tau) + sqrtf(1.f + tau * tau));
                    const float c_  = 1.f / sqrtf(1.f + t * t);
                    const float s_  = t * c_;
#pragma unroll
                    for (int k = 0; k < 4; ++k) {
                        const float gkp = G[k][pi], gkq = G[k][qi];
                        G[k][pi] = c_ * gkp - s_ * gkq;
                        G[k][qi] = s_ * gkp + c_ * gkq;
                    }
#pragma unroll
                    for (int k = 0; k < 4; ++k) {
                        const float gpk = G[pi][k], gqk = G[qi][k];
                        G[pi][k] = c_ * gpk - s_ * gqk;
                        G[qi][k] = s_ * gpk + c_ * gqk;
                    }
#pragma unroll
                    for (int k = 0; k < 4; ++k) {
                        const float vkp = V[k][pi], vkq = V[k][qi];
                        V[k][pi] = c_ * vkp - s_ * vkq;
                        V[k][qi] = s_ * vkp + c_ * vkq;
                    }
                }
            }
        }

        int   mi = 0;
        float mv = G[0][0];
#pragma unroll
        for (int k = 1; k < 4; ++k)
            if (G[k][k] < mv) { mv = G[k][k]; mi = k; }

        float hom[4];
#pragma unroll
        for (int k = 0; k < 4; ++k) hom[k] = V[k][mi];

        const float mx = fmaxf(fmaxf(fabsf(hom[0]), fabsf(hom[1])),
                               fmaxf(fabsf(hom[2]), fabsf(hom[3])));
        const float inv_mx = 1.f / mx;
#pragma unroll
        for (int k = 0; k < 4; ++k) hom[k] *= inv_mx;

        const float w    = hom[3];
        const bool  bad  = fabsf(w) < 0.005f;
        const float invw = bad ? 1.f : 1.f / w;
        const float nanv = __builtin_nanf("");

        out[pp * 3 + 0] = bad ? nanv : hom[0] * invw;
        out[pp * 3 + 1] = bad ? nanv : hom[1] * invw;
        out[pp * 3 + 2] = bad ? nanv : hom[2] * invw;
    }
}

extern "C" void kernel_launch(void* const* d_in, const int* in_sizes, int n_in,
                              void* d_out, int out_size, void* d_ws, size_t ws_size,
                              hipStream_t stream) {
    (void)in_sizes; (void)n_in; (void)d_ws; (void)ws_size; (void)out_size;
    const float* in2d  = (const float*)d_in[0];   // input_2d  (32,256,17,2)
    const float* incam = (const float*)d_in[1];   // input_cam (32,256,3,4)
    float*       out   = (float*)d_out;           // (32,1,17,3)
    // 17 blocks * 8 waves * 4 points = 544 = 32*17 points, exact cover.
    tri_dlt_kernel<<<17, 256, 0, stream>>>(in2d, incam, out);
}